// GATConv_Decoder_22316650070986
// MI455X (gfx1250) — compile-verified
//
#include <hip/hip_runtime.h>

#define N_NODES 50000
#define N_EDGES 800000
#define E_TOT   (N_EDGES + N_NODES)
#define D       128
#define HD      256   /* HEADS * D */

typedef __attribute__((ext_vector_type(2))) float v2f;
typedef __attribute__((ext_vector_type(8))) float v8f;

// ---------------------------------------------------------------------------
// h[N,256] = x[N,128] @ W[128,256] via V_WMMA_F32_16X16X4_F32 (full fp32).
// One wave computes a 16x64 strip (4 col tiles). The 16x128 A strip is
// preloaded into 32 v2f register frags (64 VGPRs), then 4 x 32 WMMAs run with
// only B-side loads in the inner loop: 128 wmma / wave, A read once per wave.
// Frag layouts (ISA 7.12.2, 32-bit):
//   A 16x4 : lanes 0-15 -> K={0,1}, lanes 16-31 -> K={2,3}
//   B 4x16 : mirrored over columns
//   C/D    : VGPR r holds M=r (lanes 0-15) / M=r+8 (lanes 16-31)
// Grid: 3125 blocks x 4 waves = 12500 waves, exact -> EXEC all-1s everywhere.
// ---------------------------------------------------------------------------
__global__ __launch_bounds__(128) void gat_gemm_wmma(const float* __restrict__ x,
                                                     const float* __restrict__ W,
                                                     float* __restrict__ h) {
  const int lane = threadIdx.x & 31;
  const int wave = threadIdx.x >> 5;
  const int tile = blockIdx.x * 4 + wave;            // 12500 strips exactly
  const int tm   = tile >> 2;                         // 3125 row tiles
  const int cg   = tile & 3;                          // 4 col groups of 64
  const int l15  = lane & 15;
  const int hi   = (lane >> 4) & 1;                   // half-wave selector

  // Preload full A strip: 32 K-step fragments held in registers.
  const float* Arow = x + (size_t)(tm * 16 + l15) * D + hi * 2;
  v2f a[32];
#pragma unroll
  for (int k = 0; k < 32; ++k) {
    a[k].x = Arow[k * 4];
    a[k].y = Arow[k * 4 + 1];
  }

  const float* Bbase = W + (size_t)(hi * 2) * HD + cg * 64 + l15;
#pragma unroll
  for (int c = 0; c < 4; ++c) {
    const float* Bcol = Bbase + c * 16;
    v8f acc = {0.f, 0.f, 0.f, 0.f, 0.f, 0.f, 0.f, 0.f};
#pragma unroll
    for (int k = 0; k < 32; ++k) {
      v2f b;
      b.x = Bcol[(size_t)(k * 4) * HD];
      b.y = Bcol[(size_t)(k * 4 + 1) * HD];
      acc = __builtin_amdgcn_wmma_f32_16x16x4_f32(false, a[k], false, b,
                                                  (short)0, acc, false, false);
    }
    float* out = h + (size_t)(tm * 16 + hi * 8) * HD + cg * 64 + c * 16 + l15;
#pragma unroll
    for (int r = 0; r < 8; ++r) out[(size_t)r * HD] = acc[r];
  }
}

// a_s[n,h] = <h[n,h,:], att_src[h,:]>, a_d likewise. One thread per (node,head).
__global__ __launch_bounds__(256) void gat_attn(const float* __restrict__ h,
                                                const float* __restrict__ att_src,
                                                const float* __restrict__ att_dst,
                                                float* __restrict__ a_s,
                                                float* __restrict__ a_d) {
  int idx = blockIdx.x * blockDim.x + threadIdx.x;
  if (idx >= N_NODES * 2) return;
  int n = idx >> 1, hh = idx & 1;
  const float* hr = h + (size_t)n * HD + hh * D;
  const float* vs = att_src + hh * D;
  const float* vd = att_dst + hh * D;
  float sa = 0.f, sd = 0.f;
#pragma unroll 4
  for (int d = 0; d < D; ++d) { float v = hr[d]; sa += v * vs[d]; sd += v * vd[d]; }
  a_s[idx] = sa;
  a_d[idx] = sd;
}

// m = -inf, s = 0, acc = 0 (per layer)
__global__ __launch_bounds__(256) void gat_init(float* __restrict__ m,
                                                float* __restrict__ s,
                                                float* __restrict__ acc) {
  int idx = blockIdx.x * blockDim.x + threadIdx.x;
  if (idx < N_NODES * D) acc[idx] = 0.f;
  if (idx < N_NODES * 2) { m[idx] = -__builtin_inff(); s[idx] = 0.f; }
}

__device__ __forceinline__ void edge_sd(int e, const long long* __restrict__ ei,
                                        int& s, int& d) {
  if (e < N_EDGES) { s = (int)ei[e]; d = (int)ei[N_EDGES + e]; }
  else             { s = d = e - N_EDGES; }                     // self loops
}

__device__ __forceinline__ float leaky(float v) { return v > 0.f ? v : 0.2f * v; }

// segment max of leaky_relu(a_s[src]+a_d[dst]) over incoming edges (native f32 atomicMax)
__global__ __launch_bounds__(256) void gat_edge_max(const long long* __restrict__ ei,
                                                    const float* __restrict__ a_s,
                                                    const float* __restrict__ a_d,
                                                    float* __restrict__ m) {
  int idx = blockIdx.x * blockDim.x + threadIdx.x;
  if (idx >= E_TOT * 2) return;
  int e = idx >> 1, hh = idx & 1;
  int si, di; edge_sd(e, ei, si, di);
  float v = leaky(a_s[si * 2 + hh] + a_d[di * 2 + hh]);
  atomicMax(&m[di * 2 + hh], v);
}

// segment sum of exp(e - m[dst]) (recompute logit instead of storing per-edge state)
__global__ __launch_bounds__(256) void gat_edge_sum(const long long* __restrict__ ei,
                                                    const float* __restrict__ a_s,
                                                    const float* __restrict__ a_d,
                                                    const float* __restrict__ m,
                                                    float* __restrict__ ssum) {
  int idx = blockIdx.x * blockDim.x + threadIdx.x;
  if (idx >= E_TOT * 2) return;
  int e = idx >> 1, hh = idx & 1;
  int si, di; edge_sd(e, ei, si, di);
  float v = leaky(a_s[si * 2 + hh] + a_d[di * 2 + hh]);
  atomicAdd(&ssum[di * 2 + hh], __expf(v - m[di * 2 + hh]));
}

// acc[dst,d] += 0.5*(alpha0*h[src,d] + alpha1*h[src,128+d])  (head-mean fused)
// One wave per edge, float4 gathers, 4 atomicAdds per lane (128/edge).
__global__ __launch_bounds__(256) void gat_edge_msg(const long long* __restrict__ ei,
                                                    const float* __restrict__ a_s,
                                                    const float* __restrict__ a_d,
                                                    const float* __restrict__ m,
                                                    const float* __restrict__ ssum,
                                                    const float* __restrict__ h,
                                                    float* __restrict__ acc) {
  int lane = threadIdx.x & 31;
  int e = blockIdx.x * 8 + (threadIdx.x >> 5);
  if (e >= E_TOT) return;
  int si, di; edge_sd(e, ei, si, di);
  float w[2];
#pragma unroll
  for (int hh = 0; hh < 2; ++hh) {
    float v = leaky(a_s[si * 2 + hh] + a_d[di * 2 + hh]);
    w[hh] = 0.5f * __expf(v - m[di * 2 + hh]) / ssum[di * 2 + hh];
  }
  const float* hr = h + (size_t)si * HD;
  int dbase = lane * 4;
  float4 u0 = *(const float4*)(hr + dbase);
  float4 u1 = *(const float4*)(hr + D + dbase);
  float* out = acc + (size_t)di * D + dbase;
  atomicAdd(out + 0, w[0] * u0.x + w[1] * u1.x);
  atomicAdd(out + 1, w[0] * u0.y + w[1] * u1.y);
  atomicAdd(out + 2, w[0] * u0.z + w[1] * u1.z);
  atomicAdd(out + 3, w[0] * u0.w + w[1] * u1.w);
}

__global__ __launch_bounds__(256) void gat_finalize(const float* __restrict__ acc,
                                                    const float* __restrict__ bias,
                                                    float* __restrict__ out) {
  int idx = blockIdx.x * blockDim.x + threadIdx.x;
  if (idx >= N_NODES * D) return;
  out[idx] = acc[idx] + bias[idx & (D - 1)];
}

extern "C" void kernel_launch(void* const* d_in, const int* in_sizes, int n_in,
                              void* d_out, int out_size, void* d_ws, size_t ws_size,
                              hipStream_t stream) {
  const float*     x   = (const float*)d_in[0];
  const long long* ei  = (const long long*)d_in[1];   // int64 edge_index [2, E]
  const float*     W1  = (const float*)d_in[2];
  const float*     as1 = (const float*)d_in[3];
  const float*     ad1 = (const float*)d_in[4];
  const float*     b1  = (const float*)d_in[5];
  const float*     W2  = (const float*)d_in[6];
  const float*     as2 = (const float*)d_in[7];
  const float*     ad2 = (const float*)d_in[8];
  const float*     b2  = (const float*)d_in[9];

  float* ws   = (float*)d_ws;
  float* h    = ws;                                   // N*256
  float* a_s  = h    + (size_t)N_NODES * HD;          // N*2
  float* a_d  = a_s  + N_NODES * 2;                   // N*2
  float* m    = a_d  + N_NODES * 2;                   // N*2
  float* ssum = m    + N_NODES * 2;                   // N*2
  float* acc  = ssum + N_NODES * 2;                   // N*128
  float* x2   = acc  + (size_t)N_NODES * D;           // N*128

  const int gemm_blocks = (N_NODES / 16) * 4 / 4;             // 3125, 4 waves each
  const int nh_blocks   = (N_NODES * 2 + 255) / 256;
  const int nd_blocks   = (N_NODES * D + 255) / 256;
  const int eh_blocks   = (E_TOT * 2 + 255) / 256;
  const int em_blocks   = (E_TOT + 7) / 8;                    // 8 waves/block

  auto layer = [&](const float* xin, const float* W, const float* as,
                   const float* ad, const float* b, float* xout) {
    gat_gemm_wmma<<<gemm_blocks, 128, 0, stream>>>(xin, W, h);
    gat_attn     <<<nh_blocks,   256, 0, stream>>>(h, as, ad, a_s, a_d);
    gat_init     <<<nd_blocks,   256, 0, stream>>>(m, ssum, acc);
    gat_edge_max <<<eh_blocks,   256, 0, stream>>>(ei, a_s, a_d, m);
    gat_edge_sum <<<eh_blocks,   256, 0, stream>>>(ei, a_s, a_d, m, ssum);
    gat_edge_msg <<<em_blocks,   256, 0, stream>>>(ei, a_s, a_d, m, ssum, h, acc);
    gat_finalize <<<nd_blocks,   256, 0, stream>>>(acc, b, xout);
  };

  layer(x,  W1, as1, ad1, b1, x2);                    // layer 1 -> x2 in ws
  layer(x2, W2, as2, ad2, b2, (float*)d_out);         // layer 2 -> d_out
}